// graph_55070070669784
// MI455X (gfx1250) — compile-verified
//
#include <hip/hip_runtime.h>

typedef float v2f __attribute__((ext_vector_type(2)));
typedef float v4f __attribute__((ext_vector_type(4)));
typedef float v8f __attribute__((ext_vector_type(8)));

#define NGR      32
#define BS       64
#define NN       2048
#define DIM      256
#define NCELLS   1024
#define ROWP     260   // LDS row stride in dwords: 256 + 4 pad -> conflict-free C reads

__global__ __launch_bounds__(256)
void fused_embed_gather_kernel(const float* __restrict__ memory,
                               const float* __restrict__ edges,
                               const int*   __restrict__ mask,
                               const float* __restrict__ W,
                               const float* __restrict__ bias,
                               float*       __restrict__ out)
{
    __shared__ float sGather[16 * ROWP];   // gathered memory rows (zeroed if masked)
    __shared__ float sOut[16 * ROWP];      // WMMA result staging for coalesced stores
    __shared__ float sX[16], sY[16];
    __shared__ int   sCell[16], sMask[16];
    __shared__ float sBias[DIM];

    const int tid = threadIdx.x;
    const int b   = blockIdx.x >> 7;            // 128 node-tiles per batch element
    const int n0  = (blockIdx.x & 127) << 4;    // first node of this 16-node tile

    // ---- stage 0: bias -> LDS; per-node meta (cell index, mask, coords)
    sBias[tid] = bias[tid];
    if (tid < 16) {
        const int n   = n0 + tid;
        const float x = edges[((size_t)b * NN + n) * 2 + 0];
        const float y = edges[((size_t)b * NN + n) * 2 + 1];
        const int gx  = ((int)floorf(x * 511.0f)) >> 4;   // // GRID_SZ(16)
        const int gy  = ((int)floorf(y * 511.0f)) >> 4;
        sX[tid]    = x;
        sY[tid]    = y;
        sCell[tid] = gx * NGR + gy;
        sMask[tid] = mask[(size_t)b * NN + n];
    }
    __syncthreads();

    // ---- stage 1: async DMA gather of 16 memory rows (1KB each) into LDS.
    // Masked rows: skip the global traffic entirely, write zeros to LDS.
    {
        const int m = tid >> 4;                 // node-in-tile 0..15
        const int j = tid & 15;                 // lane within 16-thread row group
        float* ldsrow = &sGather[m * ROWP];
        if (sMask[m]) {
#pragma unroll
            for (int p = 0; p < 4; ++p)
                *(v4f*)(ldsrow + (p * 16 + j) * 4) = (v4f){0.f, 0.f, 0.f, 0.f};
        } else {
            const unsigned lbase =
                (unsigned)(unsigned long long)ldsrow;     // LDS byte offset
            const float* row = memory + ((size_t)b * NCELLS + sCell[m]) * DIM;
#pragma unroll
            for (int p = 0; p < 4; ++p) {
                const int f4 = p * 16 + j;                // float4 index in row
                unsigned           laddr = lbase + (unsigned)(f4 * 16);
                unsigned long long gaddr = (unsigned long long)(row + f4 * 4);
                asm volatile("global_load_async_to_lds_b128 %0, %1, off"
                             :: "v"(laddr), "v"(gaddr) : "memory");
            }
        }
    }
    asm volatile("s_wait_asynccnt 0" ::: "memory");
    __syncthreads();

    // ---- stage 2: D = A(edge) x B(W) + C(bias + gathered)  via v_wmma_f32_16x16x4_f32
    const int lane = tid & 31;
    const int wave = tid >> 5;                  // 8 waves -> dims 32*wave .. 32*wave+31
    const int half = lane >> 4;                 // 0: K=0,1 side  1: K=2,3 zero pad
    const int ln   = lane & 15;

    // A matrix 16x4: lanes 0-15 hold row M=ln, K=0,1; lanes 16-31 K=2,3 (zeros)
    v2f a;
    a.x = half ? 0.f : sX[ln];
    a.y = half ? 0.f : sY[ln];

#pragma unroll
    for (int t = 0; t < 2; ++t) {
        const int d0 = wave * 32 + t * 16;
        const int d  = d0 + ln;

        // Unconditional coalesced loads, then cndmask the K=2,3 half to zero
        const float w0 = W[d * 2 + 0];
        const float w1 = W[d * 2 + 1];
        v2f bm;
        bm.x = half ? 0.f : w0;
        bm.y = half ? 0.f : w1;

        const float bv = sBias[d];
        v8f c;
#pragma unroll
        for (int v = 0; v < 8; ++v) {
            const int m = v + half * 8;         // C/D layout: vgpr v -> row m (or m+8)
            c[v] = bv + sGather[m * ROWP + d];
        }

        c = __builtin_amdgcn_wmma_f32_16x16x4_f32(false, a, false, bm,
                                                  (short)0, c, false, false);

#pragma unroll
        for (int v = 0; v < 8; ++v) {
            const int m = v + half * 8;
            sOut[m * ROWP + d] = c[v];
        }
    }
    __syncthreads();

    // ---- stage 3: coalesced non-temporal b128 stores (write-once stream;
    // keep L2 free for the re-gathered memory tensor)
    {
        const int m = tid >> 4;
        const int j = tid & 15;
        float* orow = out + ((size_t)b * NN + n0 + m) * DIM;
#pragma unroll
        for (int p = 0; p < 4; ++p) {
            const int f4 = p * 16 + j;
            v4f v = *(const v4f*)&sOut[m * ROWP + f4 * 4];
            __builtin_nontemporal_store(v, (v4f*)orow + f4);
        }
    }
}

extern "C" void kernel_launch(void* const* d_in, const int* in_sizes, int n_in,
                              void* d_out, int out_size, void* d_ws, size_t ws_size,
                              hipStream_t stream) {
    (void)in_sizes; (void)n_in; (void)out_size; (void)d_ws; (void)ws_size;
    const float* memory = (const float*)d_in[0];   // [64,1024,256] f32
    const float* edges  = (const float*)d_in[1];   // [64,2048,2]   f32
    const int*   mask   = (const int*)  d_in[2];   // [64,2048]     int (bool)
    const float* W      = (const float*)d_in[3];   // [256,2]       f32
    const float* bias   = (const float*)d_in[4];   // [256]         f32
    float*       out    = (float*)d_out;           // [64,2048,256] f32

    const int blocks = BS * (NN / 16);             // 8192 blocks, 256 thr each
    fused_embed_gather_kernel<<<blocks, 256, 0, stream>>>(memory, edges, mask,
                                                          W, bias, out);
}